// BiLSTM_CRF_81131932221674
// MI455X (gfx1250) — compile-verified
//
#include <hip/hip_runtime.h>
#include <hip/hip_bf16.h>
#include <math.h>

#define L_SEQ  16384
#define E_DIM  256
#define H_DIM  64
#define G4H    256      // 4*H
#define T_TAG  6
#define TAG_START 4
#define TAG_STOP  5
#define NEG_INF  (-10000.0f)

typedef __attribute__((ext_vector_type(2))) float v2f;
typedef __attribute__((ext_vector_type(8))) float v8f;

// ---------------------------------------------------------------------------
// Kernel 1: embedding gather + input-to-gate projection for BOTH directions.
// xg_d[l][g] = sum_k emb[sentence[l]][k] * Wih_d[g][k] + bih_d[g] + bhh_d[g]
// One block = 16 tokens (one WMMA M-tile). 128 threads = 4 waves.
// 32 output tiles of 16x16 (2 dirs x 16 N-tiles), 8 tiles per wave,
// each tile accumulated with 64 x V_WMMA_F32_16X16X4_F32 (K = 256).
// ---------------------------------------------------------------------------
__global__ __launch_bounds__(128)
void bilstm_proj_kernel(const int* __restrict__ sent,
                        const float* __restrict__ emb,
                        const float* __restrict__ Wih_f,
                        const float* __restrict__ bih_f,
                        const float* __restrict__ bhh_f,
                        const float* __restrict__ Wih_b,
                        const float* __restrict__ bih_b,
                        const float* __restrict__ bhh_b,
                        float* __restrict__ xg_f,
                        float* __restrict__ xg_b)
{
    __shared__ float xs[16][E_DIM + 4];   // +4 pad: conflict-free A reads
    const int row0 = blockIdx.x * 16;
    const int tid  = threadIdx.x;

    // Cooperative gather of 16 embedding rows (float4 vectorized, no divergence:
    // 16*64 = 1024 float4 over 128 threads = exactly 8 iters each).
    for (int i = tid; i < 16 * (E_DIM / 4); i += 128) {
        const int r  = i >> 6;            // / 64
        const int c4 = i & 63;            // % 64
        const float4 v =
            ((const float4*)(emb + (size_t)sent[row0 + r] * E_DIM))[c4];
        xs[r][c4 * 4 + 0] = v.x;
        xs[r][c4 * 4 + 1] = v.y;
        xs[r][c4 * 4 + 2] = v.z;
        xs[r][c4 * 4 + 3] = v.w;
    }
    __syncthreads();

    const int wave = tid >> 5;
    const int lane = tid & 31;
    const int m    = lane & 15;           // A: row M      / B,D: col N
    const int kh   = lane >> 4;           // K-half select within each K=4 step

    for (int t8 = 0; t8 < 8; ++t8) {
        const int tile  = wave * 8 + t8;      // 0..31
        const int dir   = tile >> 4;          // 0 = fwd, 1 = bwd
        const int ntile = tile & 15;
        const float* __restrict__ W  = dir ? Wih_b : Wih_f;
        const float* __restrict__ b1 = dir ? bih_b : bih_f;
        const float* __restrict__ b2 = dir ? bhh_b : bhh_f;
        float* __restrict__ out      = dir ? xg_b : xg_f;

        const int ncol = ntile * 16 + m;      // this lane's output column
        const float* wrow = W + (size_t)ncol * E_DIM;

        v8f acc = {};
        #pragma unroll 8
        for (int ks = 0; ks < E_DIM / 4; ++ks) {
            const int k0 = ks * 4 + kh * 2;
            // A fragment: 16x4 f32, lane m = row, (kh,elem) = K
            v2f a = { xs[m][k0], xs[m][k0 + 1] };
            // B fragment: 4x16 f32, lane m = col N, (kh,elem) = K; B[k][n]=Wih[n][k]
            v2f b = { wrow[k0], wrow[k0 + 1] };
            acc = __builtin_amdgcn_wmma_f32_16x16x4_f32(
                false, a, false, b, (short)0, acc, false, false);
        }

        const float bias = b1[ncol] + b2[ncol];
        #pragma unroll
        for (int r = 0; r < 8; ++r) {
            const int mrow = r + 8 * kh;      // D layout: VGPR r -> M = r (+8)
            out[(size_t)(row0 + mrow) * G4H + ncol] = acc[r] + bias;
        }
    }
}

// ---------------------------------------------------------------------------
// Kernel 2: the sequential LSTM scan. blockIdx.x = direction (0 fwd, 1 bwd).
// 256 threads; thread t owns gate-row t of Whh (64 floats in VGPRs).
// h, c, g live in LDS. Backward direction walks l from L-1 down to 0 and
// writes hs_b[l] in place (== reversed-scan-then-reverse of the reference).
// ---------------------------------------------------------------------------
__global__ __launch_bounds__(256)
void bilstm_scan_kernel(const float* __restrict__ xg_f,
                        const float* __restrict__ xg_b,
                        const float* __restrict__ Whh_f,
                        const float* __restrict__ Whh_b,
                        const float* __restrict__ h0,
                        const float* __restrict__ c0,
                        float* __restrict__ hs_f,
                        float* __restrict__ hs_b)
{
    __shared__ float hsd[H_DIM];
    __shared__ float csd[H_DIM];
    __shared__ float gsd[G4H];

    const int dir = blockIdx.x;
    const float* __restrict__ Whh = dir ? Whh_b : Whh_f;
    const float* __restrict__ xg  = dir ? xg_b  : xg_f;
    float* __restrict__ hs        = dir ? hs_b  : hs_f;
    const int t = threadIdx.x;

    // Private copy of this thread's Whh row (64 VGPRs).
    float wrow[H_DIM];
    #pragma unroll
    for (int k = 0; k < H_DIM; ++k) wrow[k] = Whh[(size_t)t * H_DIM + k];

    if (t < H_DIM) {
        hsd[t] = h0[dir * H_DIM + t];
        csd[t] = c0[dir * H_DIM + t];
    }
    __syncthreads();

    const int lstep = dir ? -1 : 1;
    int l = dir ? (L_SEQ - 1) : 0;

    for (int it = 0; it < L_SEQ; ++it, l += lstep) {
        // gfx1250 global_prefetch_b8 on next step's xg row (hides the only
        // global load latency inside the serial scan).
        if (it + 1 < L_SEQ)
            __builtin_prefetch(&xg[(size_t)(l + lstep) * G4H + t], 0, 0);

        float g = xg[(size_t)l * G4H + t];
        #pragma unroll
        for (int k = 0; k < H_DIM; ++k) g += wrow[k] * hsd[k];
        gsd[t] = g;
        __syncthreads();

        if (t < H_DIM) {
            const float ig = 1.0f / (1.0f + __expf(-gsd[t]));
            const float fg = 1.0f / (1.0f + __expf(-gsd[H_DIM + t]));
            const float gg = tanhf(gsd[2 * H_DIM + t]);
            const float og = 1.0f / (1.0f + __expf(-gsd[3 * H_DIM + t]));
            const float c  = fg * csd[t] + ig * gg;
            const float h  = og * tanhf(c);
            csd[t] = c;
            hsd[t] = h;
            hs[(size_t)l * H_DIM + t] = h;
        }
        __syncthreads();
    }
}

// ---------------------------------------------------------------------------
// Kernel 3: tag features. feats[l][t] = [hf[l],hb[l]] . W_h2t[t] + b_h2t[t]
// ---------------------------------------------------------------------------
__global__ __launch_bounds__(256)
void bilstm_feats_kernel(const float* __restrict__ hs_f,
                         const float* __restrict__ hs_b,
                         const float* __restrict__ W_h2t,
                         const float* __restrict__ b_h2t,
                         float* __restrict__ feats)
{
    const int idx = blockIdx.x * 256 + threadIdx.x;
    if (idx >= L_SEQ * T_TAG) return;
    const int l = idx / T_TAG;
    const int t = idx % T_TAG;
    const float* __restrict__ wf = W_h2t + t * 2 * H_DIM;
    float acc = b_h2t[t];
    #pragma unroll 8
    for (int k = 0; k < H_DIM; ++k) acc += hs_f[(size_t)l * H_DIM + k] * wf[k];
    #pragma unroll 8
    for (int k = 0; k < H_DIM; ++k) acc += hs_b[(size_t)l * H_DIM + k] * wf[H_DIM + k];
    feats[idx] = acc;
}

// ---------------------------------------------------------------------------
// Kernel 4: Viterbi forward + backtrace. One wave; lanes 0..5 = next-tag.
// bp stored as bytes in workspace. Lane 0 does the backtrace and writes
// d_out = [score, path(16384 floats)].
// ---------------------------------------------------------------------------
__global__ __launch_bounds__(32)
void bilstm_viterbi_kernel(const float* __restrict__ feats,
                           const float* __restrict__ trans,
                           unsigned char* __restrict__ bp,
                           float* __restrict__ out)
{
    __shared__ float fv[T_TAG];
    __shared__ float fv2[T_TAG];
    const int t = threadIdx.x;

    float tr[T_TAG];
    if (t < T_TAG) {
        #pragma unroll
        for (int p = 0; p < T_TAG; ++p) tr[p] = trans[t * T_TAG + p];
        fv[t] = (t == TAG_START) ? 0.0f : NEG_INF;
    }
    __syncthreads();

    for (int l = 0; l < L_SEQ; ++l) {
        if (t < T_TAG) {
            float best = fv[0] + tr[0];
            int bi = 0;
            #pragma unroll
            for (int p = 1; p < T_TAG; ++p) {
                const float s = fv[p] + tr[p];
                if (s > best) { best = s; bi = p; }   // first-max, like argmax
            }
            fv2[t] = best + feats[(size_t)l * T_TAG + t];
            bp[(size_t)l * T_TAG + t] = (unsigned char)bi;
        }
        __syncthreads();
        if (t < T_TAG) fv[t] = fv2[t];
        __syncthreads();
    }

    if (t == 0) {
        float best = -3.4e38f;
        int bt = 0;
        for (int p = 0; p < T_TAG; ++p) {
            const float s = fv[p] + trans[TAG_STOP * T_TAG + p];
            if (s > best) { best = s; bt = p; }
        }
        out[0] = best;                       // score
        int tag = bt;
        for (int l = L_SEQ - 1; l >= 0; --l) {
            out[1 + l] = (float)tag;         // path[l]
            tag = (int)bp[(size_t)l * T_TAG + tag];
        }
    }
}

// ---------------------------------------------------------------------------
extern "C" void kernel_launch(void* const* d_in, const int* in_sizes, int n_in,
                              void* d_out, int out_size, void* d_ws, size_t ws_size,
                              hipStream_t stream)
{
    (void)in_sizes; (void)n_in; (void)out_size; (void)ws_size;

    const int*   sent   = (const int*)  d_in[0];
    const float* emb    = (const float*)d_in[1];
    const float* Wih_f  = (const float*)d_in[2];
    const float* Whh_f  = (const float*)d_in[3];
    const float* bih_f  = (const float*)d_in[4];
    const float* bhh_f  = (const float*)d_in[5];
    const float* Wih_b  = (const float*)d_in[6];
    const float* Whh_b  = (const float*)d_in[7];
    const float* bih_b  = (const float*)d_in[8];
    const float* bhh_b  = (const float*)d_in[9];
    const float* W_h2t  = (const float*)d_in[10];
    const float* b_h2t  = (const float*)d_in[11];
    const float* trans  = (const float*)d_in[12];
    const float* h0     = (const float*)d_in[13];
    const float* c0     = (const float*)d_in[14];
    float* out = (float*)d_out;

    // Workspace carve (≈ 42.4 MB).
    float* xg_f  = (float*)d_ws;
    float* xg_b  = xg_f + (size_t)L_SEQ * G4H;
    float* hs_f  = xg_b + (size_t)L_SEQ * G4H;
    float* hs_b  = hs_f + (size_t)L_SEQ * H_DIM;
    float* feats = hs_b + (size_t)L_SEQ * H_DIM;
    unsigned char* bp = (unsigned char*)(feats + (size_t)L_SEQ * T_TAG);

    // 1) Gather + input projections (WMMA f32): 1024 blocks of 16 tokens.
    bilstm_proj_kernel<<<L_SEQ / 16, 128, 0, stream>>>(
        sent, emb, Wih_f, bih_f, bhh_f, Wih_b, bih_b, bhh_b, xg_f, xg_b);

    // 2) Sequential scans, both directions concurrently.
    bilstm_scan_kernel<<<2, 256, 0, stream>>>(
        xg_f, xg_b, Whh_f, Whh_b, h0, c0, hs_f, hs_b);

    // 3) Tag features.
    bilstm_feats_kernel<<<(L_SEQ * T_TAG + 255) / 256, 256, 0, stream>>>(
        hs_f, hs_b, W_h2t, b_h2t, feats);

    // 4) Viterbi + backtrace.
    bilstm_viterbi_kernel<<<1, 32, 0, stream>>>(feats, trans, bp, out);
}